// CavityModel_67671504716163
// MI455X (gfx1250) — compile-verified
//
#include <hip/hip_runtime.h>
#include <hip/hip_bf16.h>

typedef float v2f __attribute__((ext_vector_type(2)));
typedef float v8f __attribute__((ext_vector_type(8)));

#define NENV 256
#define NTYPES 6
#define GD 18
#define GPTS (GD*GD*GD)          // 5832
#define SLOPE 0.01f
#define BN_EPS 1e-5f

#define WMMA_F32(acc, a, b) \
    __builtin_amdgcn_wmma_f32_16x16x4_f32(false, (a), false, (b), (short)0, (acc), false, false)

// ---------------------------------------------------------------------------
// 1. Gaussian blurring: one block per (env, type) channel. Separable Gaussian:
//    per-atom 54 exp() table in LDS, field accumulated atomic-free.
// ---------------------------------------------------------------------------
__global__ void blur_kernel(const float* __restrict__ pos,
                            const int* __restrict__ types,
                            float* __restrict__ fields) {
    __shared__ int   sIdx[100];
    __shared__ int   sCount;
    __shared__ float sTab[100][54];   // [atom][axis*18+bin]
    __shared__ float sInv[100];
    const int b = blockIdx.x / NTYPES;
    const int t = blockIdx.x % NTYPES;
    const int tid = threadIdx.x;

    if (tid == 0) {
        int c = 0, base = b * 100;
        for (int i = 0; i < 100; ++i)
            if (types[base + i] == t) sIdx[c++] = base + i;
        sCount = c;
    }
    __syncthreads();
    const int na = sCount;

    for (int e = tid; e < na * 54; e += 256) {
        int a = e / 54, r = e % 54, axis = r / 18, bin = r % 18;
        float coord = pos[sIdx[a] * 3 + axis];
        float d = (float)bin - 8.5f - coord;
        sTab[a][r] = __expf(-d * d * (1.0f / 0.72f));   // 2*sigma_p^2 = 0.72
    }
    __syncthreads();
    for (int a = tid; a < na; a += 256) {
        float sx = 0.f, sy = 0.f, sz = 0.f;
        for (int i = 0; i < 18; ++i) {
            sx += sTab[a][i]; sy += sTab[a][18 + i]; sz += sTab[a][36 + i];
        }
        sInv[a] = 1.0f / (sx * sy * sz);
    }
    __syncthreads();

    float* outp = fields + (size_t)(b * NTYPES + t) * GPTS;
    for (int g = tid; g < GPTS; g += 256) {
        int ix = g / 324, iy = (g / 18) % 18, iz = g % 18;
        float acc = 0.f;
        for (int a = 0; a < na; ++a)
            acc += sTab[a][ix] * sTab[a][18 + iy] * sTab[a][36 + iz] * sInv[a];
        outp[g] = acc;
    }
}

// ---------------------------------------------------------------------------
// 2a. conv1 via WMMA implicit GEMM. One block (8 waves) per env.
//     Dynamic LDS: 6 zero-padded 20^3 channels + weights [6][16][28] (tap 27
//     zero) = 50688 floats = 202752 B.  Wave tile: 16 pos x 16 out-ch.
//     Channel loop is ROLLED (pointer bumps) + inner 7 K-groups unrolled with
//     14 statically-indexed per-lane tap offsets -> no scratch spills.
//     Epilogue pools along w in-register -> wp [256][16][18][18][9].
// ---------------------------------------------------------------------------
__global__ void __launch_bounds__(256, 1)
conv1_wmma_kernel(const float* __restrict__ fields,
                  const float* __restrict__ w,    // [16][6][27]
                  float* __restrict__ wp) {
    extern __shared__ float smem[];
    float* sPad = smem;           // 6 * 8000 (padded 20^3 per channel)
    float* sWt  = smem + 48000;   // [c][n][28], col 27 = 0
    const int b = blockIdx.x;
    const int tid = threadIdx.x;
    const int wave = tid >> 5, lane = tid & 31;
    const int half = lane >> 4, lm = lane & 15;

    for (int i = tid; i < 48000; i += 256) sPad[i] = 0.f;
    __syncthreads();
    for (int i = tid; i < 6 * GPTS; i += 256) {
        int c = i / GPTS, g = i % GPTS;
        int z = g / 324, y = (g / 18) % 18, x = g % 18;
        sPad[c * 8000 + (z + 1) * 400 + (y + 1) * 20 + (x + 1)] =
            fields[(size_t)(b * 6 + c) * GPTS + g];
    }
    for (int i = tid; i < 6 * 16 * 28; i += 256) {
        int k = i % 28, cn = i / 28, n = cn % 16, c = cn / 16;
        sWt[i] = (k < 27) ? w[(n * 6 + c) * 27 + k] : 0.f;
    }
    __syncthreads();

    // per-lane tap offsets for this lane-half: taps (4kk+2h, 4kk+2h+1)
    int toff[14];
#pragma unroll
    for (int kk = 0; kk < 7; ++kk) {
        int t0 = kk * 4 + 2 * half;
        toff[2 * kk]     = (t0 / 9) * 400 + ((t0 / 3) % 3) * 20 + (t0 % 3);
        int t1 = t0 + 1;
        toff[2 * kk + 1] = (t1 / 9) * 400 + ((t1 / 3) % 3) * 20 + (t1 % 3);
    }

    for (int tile = wave; tile < 365; tile += 8) {       // ceil(5832/16)
        int pos = tile * 16 + lm;
        int posc = pos < GPTS ? pos : GPTS - 1;
        int z = posc / 324, y = (posc / 18) % 18, x = posc % 18;
        int pbase = z * 400 + y * 20 + x;                // tap(kd,kh,kw) -> +kd*400+kh*20+kw
        v8f acc = {};
        const float* pA = sPad + pbase;
        const float* pB = sWt + lm * 28 + 2 * half;
#pragma unroll 1
        for (int c = 0; c < 6; ++c) {
#pragma unroll
            for (int kk = 0; kk < 7; ++kk) {
                v2f a, bf;
                a[0] = pA[toff[2 * kk]];
                a[1] = pA[toff[2 * kk + 1]];
                float2 bb = *(const float2*)(pB + 4 * kk);
                bf[0] = bb.x; bf[1] = bb.y;
                acc = WMMA_F32(acc, a, bf);
            }
            pA += 8000;
            pB += 16 * 28;
        }
        // in-register w-pool: rows (m, m+1) = acc[v], acc[v+1], v even
#pragma unroll
        for (int v = 0; v < 8; v += 2) {
            int p0 = tile * 16 + v + 8 * half;           // even position
            if (p0 < GPTS) {
                float mx = fmaxf(acc[v], acc[v + 1]);
                int zz = p0 / 324, yy = (p0 / 18) % 18, px = (p0 % 18) >> 1;
                wp[((size_t)(b * 16 + lm)) * 2916 + zz * 162 + yy * 9 + px] = mx;
            }
        }
    }
}

// ---------------------------------------------------------------------------
// 2b. finish conv1 pooling (z,y pairs) + bias + BN stats -> y1 [256,16,729]
// ---------------------------------------------------------------------------
__global__ void pool1_kernel(const float* __restrict__ wp,
                             const float* __restrict__ bias,
                             float* __restrict__ y1,
                             float* __restrict__ sum, float* __restrict__ sq) {
    __shared__ float sRed[256];
    const int b = blockIdx.x >> 4, co = blockIdx.x & 15, tid = threadIdx.x;
    const float* src = wp + (size_t)(b * 16 + co) * 2916;
    const float bv = bias[co];
    float ls = 0.f, ls2 = 0.f;
    for (int cell = 0; cell < 3; ++cell) {
        int p = tid + cell * 256;
        if (p >= 729) break;
        int pz = p / 81, py = (p / 9) % 9, px = p % 9;
        float m = fmaxf(fmaxf(src[(2 * pz) * 162 + (2 * py) * 9 + px],
                              src[(2 * pz) * 162 + (2 * py + 1) * 9 + px]),
                        fmaxf(src[(2 * pz + 1) * 162 + (2 * py) * 9 + px],
                              src[(2 * pz + 1) * 162 + (2 * py + 1) * 9 + px]));
        m += bv;
        y1[((size_t)(b * 16 + co)) * 729 + p] = m;
        ls += m; ls2 += m * m;
    }
    sRed[tid] = ls; __syncthreads();
    for (int s = 128; s > 0; s >>= 1) { if (tid < s) sRed[tid] += sRed[tid + s]; __syncthreads(); }
    if (tid == 0) atomicAdd(&sum[co], sRed[0]);
    __syncthreads();
    sRed[tid] = ls2; __syncthreads();
    for (int s = 128; s > 0; s >>= 1) { if (tid < s) sRed[tid] += sRed[tid + s]; __syncthreads(); }
    if (tid == 0) atomicAdd(&sq[co], sRed[0]);
}

// ---------------------------------------------------------------------------
// BatchNorm (batch stats) + LeakyReLU, elementwise in place.
// ---------------------------------------------------------------------------
__global__ void bn_leaky_kernel(float* __restrict__ x, int total, int chanDiv, int nchan,
                                const float* __restrict__ g, const float* __restrict__ bta,
                                const float* __restrict__ sum, const float* __restrict__ sq,
                                float invCnt) {
    int idx = blockIdx.x * 256 + threadIdx.x;
    if (idx >= total) return;
    int co = (idx / chanDiv) % nchan;
    float m = sum[co] * invCnt;
    float v = sq[co] * invCnt - m * m;
    float val = (x[idx] - m) * rsqrtf(v + BN_EPS) * g[co] + bta[co];
    x[idx] = val >= 0.f ? val : SLOPE * val;
}

// ---------------------------------------------------------------------------
// 3. conv2 via WMMA implicit GEMM (valid conv: no halo). One block per env.
//    K = 16 channels x 28 taps (tap 27 = zero weight col).  ci loop ROLLED,
//    7 K-groups unrolled with 14 static per-lane offsets (same as conv1).
//    Dynamic LDS: sIn 11664 + sW 32*448 + sConv 343*32 + sPool 864
//    = 37840 floats = 151360 B.
// ---------------------------------------------------------------------------
__global__ void __launch_bounds__(256, 1)
conv2_wmma_kernel(const float* __restrict__ y1,
                  const float* __restrict__ w,    // [32][16][27]
                  const float* __restrict__ bias,
                  float* __restrict__ y2,         // [256][32][27]
                  float* __restrict__ sum, float* __restrict__ sq) {
    extern __shared__ float smem[];
    float* sIn   = smem;                   // 11664
    float* sW    = smem + 11664;           // [n][ci][28] = 14336
    float* sConv = smem + 11664 + 14336;   // 343*32 = 10976
    float* sPool = sConv + 10976;          // 864
    const int b = blockIdx.x, tid = threadIdx.x;
    const int wave = tid >> 5, lane = tid & 31;
    const int half = lane >> 4, lm = lane & 15;

    for (int i = tid; i < 11664; i += 256) sIn[i] = y1[(size_t)b * 11664 + i];
    for (int i = tid; i < 32 * 448; i += 256) {
        int n = i / 448, rem = i % 448, ci = rem / 28, k = rem % 28;
        sW[i] = (k < 27) ? w[n * 432 + ci * 27 + k] : 0.f;
    }
    __syncthreads();

    int toff[14];
#pragma unroll
    for (int kk = 0; kk < 7; ++kk) {
        int t0 = kk * 4 + 2 * half;
        toff[2 * kk]     = (t0 / 9) * 81 + ((t0 / 3) % 3) * 9 + (t0 % 3);
        int t1 = t0 + 1;
        toff[2 * kk + 1] = (t1 / 9) * 81 + ((t1 / 3) % 3) * 9 + (t1 % 3);
    }

    for (int tile = wave; tile < 44; tile += 8) {        // 22 mtiles * 2 ntiles
        int mt = tile % 22, nt = tile / 22;
        int pos = mt * 16 + lm;
        int posc = pos < 343 ? pos : 342;
        int oz = posc / 49, oy = (posc / 7) % 7, ox = posc % 7;
        int n = nt * 16 + lm;
        v8f acc = {};
        const float* pA = sIn + oz * 81 + oy * 9 + ox;
        const float* pB = sW + n * 448 + 2 * half;
#pragma unroll 1
        for (int ci = 0; ci < 16; ++ci) {
#pragma unroll
            for (int kk = 0; kk < 7; ++kk) {
                v2f a, bf;
                a[0] = pA[toff[2 * kk]];
                a[1] = pA[toff[2 * kk + 1]];
                float2 bb = *(const float2*)(pB + 4 * kk);
                bf[0] = bb.x; bf[1] = bb.y;
                acc = WMMA_F32(acc, a, bf);
            }
            pA += 729;
            pB += 28;
        }
#pragma unroll
        for (int v = 0; v < 8; ++v) {
            int p = mt * 16 + v + 8 * half;
            if (p < 343) sConv[p * 32 + n] = acc[v];
        }
    }
    __syncthreads();

    for (int i = tid; i < 864; i += 256) {               // 27 cells * 32 ch
        int cell = i >> 5, co = i & 31;
        int pz = cell / 9, py = (cell / 3) % 3, px = cell % 3;
        float m = -1e30f;
#pragma unroll
        for (int sub = 0; sub < 8; ++sub) {
            int z = 2 * pz + ((sub >> 2) & 1);
            int y = 2 * py + ((sub >> 1) & 1);
            int x = 2 * px + (sub & 1);
            m = fmaxf(m, sConv[(z * 49 + y * 7 + x) * 32 + co]);
        }
        m += bias[co];
        y2[((size_t)(b * 32 + co)) * 27 + cell] = m;
        sPool[cell * 32 + co] = m;
    }
    __syncthreads();
    if (tid < 32) {
        float s = 0.f, s2 = 0.f;
        for (int c = 0; c < 27; ++c) { float v = sPool[c * 32 + tid]; s += v; s2 += v * v; }
        atomicAdd(&sum[tid], s); atomicAdd(&sq[tid], s2);
    }
}

// ---------------------------------------------------------------------------
// 4. conv3 (pad=1, 32->64, 3^3) + maxpool(2) -> [256,64]. One block per env.
// ---------------------------------------------------------------------------
__global__ void conv3_pool_kernel(const float* __restrict__ y2,
                                  const float* __restrict__ w,   // [64][32][27]
                                  const float* __restrict__ bias,
                                  float* __restrict__ y3,        // [256][64]
                                  float* __restrict__ sum, float* __restrict__ sq) {
    __shared__ float sIn[32 * 27];
    __shared__ float sConv[512];
    const int b = blockIdx.x;
    const int tid = threadIdx.x;

    for (int i = tid; i < 32 * 27; i += 256) sIn[i] = y2[(size_t)b * (32 * 27) + i];
    __syncthreads();

    for (int v = tid; v < 512; v += 256) {
        int co = v >> 3, sub = v & 7;
        int oz = (sub >> 2) & 1, oy = (sub >> 1) & 1, ox = sub & 1;
        const float* wpt = w + (size_t)co * 32 * 27;
        float s = 0.f;
        for (int ci = 0; ci < 32; ++ci)
#pragma unroll
            for (int kd = 0; kd < 3; ++kd) {
                int z = oz + kd - 1; if ((unsigned)z >= 3u) continue;
#pragma unroll
                for (int kh = 0; kh < 3; ++kh) {
                    int y = oy + kh - 1; if ((unsigned)y >= 3u) continue;
#pragma unroll
                    for (int kw = 0; kw < 3; ++kw) {
                        int x = ox + kw - 1; if ((unsigned)x >= 3u) continue;
                        s += sIn[ci * 27 + z * 9 + y * 3 + x]
                           * wpt[ci * 27 + kd * 9 + kh * 3 + kw];
                    }
                }
            }
        sConv[v] = s;
    }
    __syncthreads();
    if (tid < 64) {
        float m = sConv[tid * 8];
#pragma unroll
        for (int j = 1; j < 8; ++j) m = fmaxf(m, sConv[tid * 8 + j]);
        m += bias[tid];
        y3[(size_t)b * 64 + tid] = m;
        atomicAdd(&sum[tid], m); atomicAdd(&sq[tid], m * m);
    }
}

// ---------------------------------------------------------------------------
// 5. Dense GEMM via V_WMMA_F32_16X16X4_F32 (wave32 f32 matrix pipe).
//    out[M,N] = A[M,K] @ W[N,K]^T + bias.  For n >= N the load address is
//    CLAMPED (not zeroed): garbage in B column n only affects output column n,
//    which is never written -> branch-free inner loop, EXEC stays all-ones.
// ---------------------------------------------------------------------------
__global__ void wmma_dense_kernel(const float* __restrict__ A, int lda,
                                  const float* __restrict__ W, int ldw,
                                  const float* __restrict__ bias,
                                  float* __restrict__ out, int N, int K,
                                  int mtiles, int doLeaky, int doStats,
                                  float* __restrict__ sum, float* __restrict__ sq) {
    const int wave = threadIdx.x >> 5;
    const int lane = threadIdx.x & 31;
    const int tile = blockIdx.x * 4 + wave;
    const int mtile = tile % mtiles;
    const int ntile = tile / mtiles;
    const int half = lane >> 4;
    const int mrow = mtile * 16 + (lane & 15);
    const int n    = ntile * 16 + (lane & 15);
    const bool nok = (n < N);
    const int ncl  = nok ? n : (N - 1);

    v8f acc = {};
    const float* ap = A + (size_t)mrow * lda + half * 2;
    const float* bp = W + (size_t)ncl * ldw + half * 2;
    const int ksteps = K >> 2;
    for (int kk = 0; kk < ksteps; ++kk) {
        v2f a, bf;
        float2 av = *(const float2*)(ap + kk * 4);
        float2 bv = *(const float2*)(bp + kk * 4);
        a[0] = av.x; a[1] = av.y;
        bf[0] = bv.x; bf[1] = bv.y;
        acc = WMMA_F32(acc, a, bf);
    }
    if (!nok) return;
    float bv = bias[n];
    float s = 0.f, s2 = 0.f;
#pragma unroll
    for (int v = 0; v < 8; ++v) {
        int m = mtile * 16 + v + half * 8;
        float val = acc[v] + bv;
        if (doLeaky) val = val >= 0.f ? val : SLOPE * val;
        out[(size_t)m * N + n] = val;
        s += val; s2 += val * val;
    }
    if (doStats) { atomicAdd(&sum[n], s); atomicAdd(&sq[n], s2); }
}

// ---------------------------------------------------------------------------
// Launch
// ---------------------------------------------------------------------------
extern "C" void kernel_launch(void* const* d_in, const int* in_sizes, int n_in,
                              void* d_out, int out_size, void* d_ws, size_t ws_size,
                              hipStream_t stream) {
    const float* pos   = (const float*)d_in[0];
    const float* c1w   = (const float*)d_in[1];
    const float* c1b   = (const float*)d_in[2];
    const float* bn1g  = (const float*)d_in[3];
    const float* bn1b  = (const float*)d_in[4];
    const float* c2w   = (const float*)d_in[5];
    const float* c2b   = (const float*)d_in[6];
    const float* bn2g  = (const float*)d_in[7];
    const float* bn2b  = (const float*)d_in[8];
    const float* c3w   = (const float*)d_in[9];
    const float* c3b   = (const float*)d_in[10];
    const float* bn3g  = (const float*)d_in[11];
    const float* bn3b  = (const float*)d_in[12];
    const float* d1w   = (const float*)d_in[13];
    const float* d1b   = (const float*)d_in[14];
    const float* bn4g  = (const float*)d_in[15];
    const float* bn4b  = (const float*)d_in[16];
    const float* d2w   = (const float*)d_in[17];
    const float* d2b   = (const float*)d_in[18];
    const float* d3w   = (const float*)d_in[19];
    const float* d3b   = (const float*)d_in[20];
    const int*   types = (const int*)d_in[22];
    float*       outp  = (float*)d_out;

    // workspace layout (floats): ~96.7 MB total
    float* ws     = (float*)d_ws;
    float* fields = ws;                                   // 256*6*5832 = 8,957,952
    float* wp     = fields + (size_t)NENV * 6 * GPTS;     // 256*16*2916 = 11,943,936
    float* y1     = wp + (size_t)NENV * 16 * 2916;        // 256*16*729 = 2,985,984
    float* y2     = y1 + (size_t)NENV * 16 * 729;         // 256*32*27  =   221,184
    float* y3     = y2 + (size_t)NENV * 32 * 27;          // 256*64
    float* h1     = y3 + (size_t)NENV * 64;               // 256*128
    float* h2     = h1 + (size_t)NENV * 128;              // 256*100
    float* stats  = h2 + (size_t)NENV * 100;              // 480 floats
    float* sum1 = stats,       *sq1 = stats + 16;
    float* sum2 = stats + 32,  *sq2 = stats + 64;
    float* sum3 = stats + 96,  *sq3 = stats + 160;
    float* sum4 = stats + 224, *sq4 = stats + 352;

    hipMemsetAsync(stats, 0, 480 * sizeof(float), stream);

    // 1. gaussian blurring -> fields [256,6,18,18,18]
    blur_kernel<<<NENV * NTYPES, 256, 0, stream>>>(pos, types, fields);

    // 2. conv1 (WMMA implicit GEMM, 202752 B dynamic LDS) -> wp, then pool+stats
    conv1_wmma_kernel<<<NENV, 256, 50688 * sizeof(float), stream>>>(fields, c1w, wp);
    pool1_kernel<<<NENV * 16, 256, 0, stream>>>(wp, c1b, y1, sum1, sq1);
    bn_leaky_kernel<<<(NENV * 16 * 729) / 256, 256, 0, stream>>>(
        y1, NENV * 16 * 729, 729, 16, bn1g, bn1b, sum1, sq1, 1.0f / (NENV * 729.0f));

    // 3. conv2 (WMMA implicit GEMM, 151360 B dynamic LDS) -> y2 + stats
    conv2_wmma_kernel<<<NENV, 256, 37840 * sizeof(float), stream>>>(
        y1, c2w, c2b, y2, sum2, sq2);
    bn_leaky_kernel<<<(NENV * 32 * 27) / 256, 256, 0, stream>>>(
        y2, NENV * 32 * 27, 27, 32, bn2g, bn2b, sum2, sq2, 1.0f / (NENV * 27.0f));

    // 4. conv3 + pool -> y3 [256,64], stats3
    conv3_pool_kernel<<<NENV, 256, 0, stream>>>(y2, c3w, c3b, y3, sum3, sq3);
    bn_leaky_kernel<<<(NENV * 64) / 256, 256, 0, stream>>>(
        y3, NENV * 64, 1, 64, bn3g, bn3b, sum3, sq3, 1.0f / 256.0f);

    // 5. dense1 (WMMA): h1 = y3 @ d1w^T + d1b, stats4 ; then BN4+leaky
    wmma_dense_kernel<<<32, 128, 0, stream>>>(y3, 64, d1w, 64, d1b, h1,
                                              128, 64, 16, 0, 1, sum4, sq4);
    bn_leaky_kernel<<<(NENV * 128) / 256, 256, 0, stream>>>(
        h1, NENV * 128, 1, 128, bn4g, bn4b, sum4, sq4, 1.0f / 256.0f);

    // 6. dense2 (WMMA) + leaky: h2 = leaky(h1 @ d2w^T + d2b)
    wmma_dense_kernel<<<28, 128, 0, stream>>>(h1, 128, d2w, 128, d2b, h2,
                                              100, 128, 16, 1, 0, nullptr, nullptr);

    // 7. dense3 (WMMA): out = h2 @ d3w^T + d3b   [256,20]
    wmma_dense_kernel<<<8, 128, 0, stream>>>(h2, 100, d3w, 100, d3b, outp,
                                             20, 100, 16, 0, 0, nullptr, nullptr);
}